// PostProcessor3D_82523501625529
// MI455X (gfx1250) — compile-verified
//
#include <hip/hip_runtime.h>

// 3D NMS post-process: threshold @0.5, 9x9x9 max-pool (stride 1), keep strict
// local maxima.  Volume: D=64, H=512, W=512, f32.  Memory-bound: ~128 MiB of
// mandatory HBM traffic -> ~5.5us floor at 23.3 TB/s; halo re-reads hit L2
// (192 MB covers the whole volume).  Uses gfx1250 async global->LDS copies
// for the stage-in, separable max in LDS, depth ring buffer in LDS.

#define THRESH   0.5f
#define DD       64
#define HH       512
#define WW       512
#define RR       4
#define TT       32            // output tile edge (H and W)
#define TIN      (TT + 2*RR)   // 40: input tile edge with halo
#define ZCHUNK   16            // depths per block
#define NTHREADS 256

#define AS1 __attribute__((address_space(1)))
#define AS3 __attribute__((address_space(3)))

#if defined(__HIP_DEVICE_COMPILE__) && __has_builtin(__builtin_amdgcn_global_load_async_to_lds_b32)
#define HAVE_ASYNC 1
#else
#define HAVE_ASYNC 0
#endif

__device__ __forceinline__ void wait_async_all() {
#if defined(__HIP_DEVICE_COMPILE__) && __has_builtin(__builtin_amdgcn_s_wait_asynccnt)
    __builtin_amdgcn_s_wait_asynccnt(0);
#elif defined(__HIP_DEVICE_COMPILE__)
    asm volatile("s_wait_asynccnt 0" ::: "memory");
#endif
}

__global__ __launch_bounds__(NTHREADS) void nms3d_kernel(
    const float* __restrict__ in, float* __restrict__ out)
{
    __shared__ float s_raw[TIN * TIN];      //  6400 B: raw slice tile (with halo)
    __shared__ float s_row[TIN * TT];       //  5120 B: row-max (W window) result
    __shared__ float s_ring[9][TT * TT];    // 36864 B: 2D-pooled slices, depth ring

    const int tid = threadIdx.x;
    const int w0  = blockIdx.x * TT;
    const int h0  = blockIdx.y * TT;
    const int d0  = blockIdx.z * ZCHUNK;

    for (int z = d0 - RR; z < d0 + ZCHUNK + RR; ++z) {
        const int slot = (z + 36) % 9;      // z >= -4, so +36 keeps it non-negative

        if (z >= 0 && z < DD) {
            // ---- Phase A: stage raw 40x40 slice tile into LDS (async DMA).
            // Addresses are clamped into the volume; out-of-range entries are
            // replaced with 0 during Phase B, so clamped garbage is harmless.
            const float* slice = in + (size_t)z * (HH * WW);
            for (int i = tid; i < TIN * TIN; i += NTHREADS) {
                const int r  = i / TIN;
                const int c  = i - r * TIN;
                int gh = h0 - RR + r; gh = gh < 0 ? 0 : (gh > HH - 1 ? HH - 1 : gh);
                int gw = w0 - RR + c; gw = gw < 0 ? 0 : (gw > WW - 1 ? WW - 1 : gw);
                const float* gp = slice + (size_t)gh * WW + gw;
#if HAVE_ASYNC
                __builtin_amdgcn_global_load_async_to_lds_b32(
                    (AS1 int*)const_cast<float*>(gp),
                    (AS3 int*)&s_raw[i], /*offset=*/0, /*cpol=*/0);
#else
                s_raw[i] = *gp;
#endif
            }
#if HAVE_ASYNC
            wait_async_all();
#endif
            __syncthreads();   // also orders previous iteration's Phase D before ring write

            // ---- Phase B: max over W window (9 wide), threshold + OOB->0 on read.
            for (int i = tid; i < TIN * TT; i += NTHREADS) {
                const int r  = i >> 5;        // 0..39 (input row)
                const int c  = i & 31;        // 0..31 (output col)
                const int gh = h0 - RR + r;
                float m = 0.0f;
                if ((unsigned)gh < (unsigned)HH) {
                    const int gw = w0 - RR + c;
#pragma unroll
                    for (int j = 0; j < 9; ++j) {
                        const float v = s_raw[r * TIN + c + j];
                        const float t = ((unsigned)(gw + j) < (unsigned)WW && v > THRESH)
                                            ? v : 0.0f;
                        m = fmaxf(m, t);
                    }
                }
                s_row[i] = m;
            }
            __syncthreads();

            // ---- Phase C: max over H window (9 tall) -> depth ring slot.
            for (int i = tid; i < TT * TT; i += NTHREADS) {
                const int r = i >> 5;
                const int c = i & 31;
                float m = 0.0f;
#pragma unroll
                for (int j = 0; j < 9; ++j)
                    m = fmaxf(m, s_row[(r + j) * TT + c]);
                s_ring[slot][i] = m;
            }
            __syncthreads();
        } else {
            // OOB depth: pad ring slot with zeros (equivalent to -inf padding,
            // since every pooled value is >= 0).
            __syncthreads();   // wait for previous Phase D readers of this slot
            for (int i = tid; i < TT * TT; i += NTHREADS) s_ring[slot][i] = 0.0f;
            __syncthreads();
        }

        // ---- Phase D: emit output for depth d = z - R.  The 9-wide depth
        // window over consecutive depths covers ALL 9 ring slots exactly once,
        // so the reduction is order-independent: just max over all slots.
        const int d = z - RR;
        if (d >= d0 && d < d0 + ZCHUNK) {
            for (int i = tid; i < TT * TT; i += NTHREADS) {
                const int r = i >> 5;
                const int c = i & 31;
                float conf = 0.0f;
#pragma unroll
                for (int j = 0; j < 9; ++j)
                    conf = fmaxf(conf, s_ring[j][i]);
                const size_t gidx = ((size_t)d * HH + (size_t)(h0 + r)) * WW
                                    + (size_t)(w0 + c);
                const float v = in[gidx];            // L2-hot (loaded 4 z-steps ago)
                const float t = (v > THRESH) ? v : 0.0f;
                out[gidx] = (conf > 0.0f && conf == t) ? conf : 0.0f;
            }
        }
    }
}

extern "C" void kernel_launch(void* const* d_in, const int* in_sizes, int n_in,
                              void* d_out, int out_size, void* d_ws, size_t ws_size,
                              hipStream_t stream) {
    (void)in_sizes; (void)n_in; (void)out_size; (void)d_ws; (void)ws_size;
    const float* in  = (const float*)d_in[0];
    float*       out = (float*)d_out;
    dim3 grid(WW / TT, HH / TT, DD / ZCHUNK);   // 16 x 16 x 4 = 1024 blocks
    dim3 block(NTHREADS);                        // 8 waves (wave32)
    hipLaunchKernelGGL(nms3d_kernel, grid, block, 0, stream, in, out);
}